// biGATHeadLayer_67259187855852
// MI455X (gfx1250) — compile-verified
//
#include <hip/hip_runtime.h>
#include <hip/hip_bf16.h>
#include <math.h>

#define IN_DIM  128
#define OUT_DIM 128
#define BIG     1000000000.0f
#define EPS_BETA 1e-06f
#define EPS_BN   1e-05f

typedef __attribute__((ext_vector_type(2))) float v2f;
typedef __attribute__((ext_vector_type(8))) float v8f;

// ---------------------------------------------------------------------------
// fill kernel (graph-capture safe init; no hipMemset)
// ---------------------------------------------------------------------------
__global__ void fill_kernel(float* __restrict__ p, float v, int n) {
    int i = blockIdx.x * blockDim.x + threadIdx.x;
    if (i < n) p[i] = v;
}

// ---------------------------------------------------------------------------
// Kernel 1: z = h @ W + b, inf->BIG clamp.  fp32 WMMA 16x16x4.
// One wave computes one 16x16 tile of z; 8 waves/block cover N=128.
// A layout (16x4 f32): lanes 0-15 hold M=0..15; VGPR0 = K{0|2}, VGPR1 = K{1|3}
//   (low/high lane half selects K-pair).
// B layout (4x16 f32): N = lane&15 across lanes; K across VGPRs/halves (mirror of A).
// C/D layout: VGPR i -> M = i + 8*(lane>=16), N = lane&15.
// Fast path (full tile, n % 16 == 0 case): branch-free loads and 8 constant-
// offset stores so the compiler can clause them; guarded path only for a
// partial tail tile.
// ---------------------------------------------------------------------------
__global__ __launch_bounds__(256) void gemm_bias_kernel(
    const float* __restrict__ h, const float* __restrict__ W,
    const float* __restrict__ bias, float* __restrict__ z, int n)
{
    const int lane  = threadIdx.x & 31;
    const int wave  = threadIdx.x >> 5;
    const int row0  = blockIdx.x * 16;
    const int col0  = wave * 16;
    const int lh    = lane & 15;
    const int hsel  = lane >> 4;          // 0 = lanes 0-15, 1 = lanes 16-31
    const int kh    = hsel * 2;           // K offset for this lane half

    const bool full = (row0 + 16 <= n);
    const int rowA  = full ? (row0 + lh) : min(row0 + lh, n - 1);
    const float* Arow = h + (size_t)rowA * IN_DIM;
    const float* Bcol = W + col0 + lh;

    v8f acc = {};
    #pragma unroll 8
    for (int k = 0; k < IN_DIM; k += 4) {
        v2f a, b;
        a.x = Arow[k + kh];
        a.y = Arow[k + kh + 1];
        b.x = Bcol[(size_t)(k + kh) * OUT_DIM];
        b.y = Bcol[(size_t)(k + kh + 1) * OUT_DIM];
        acc = __builtin_amdgcn_wmma_f32_16x16x4_f32(
                  false, a, false, b, (short)0, acc, false, false);
    }

    const int col = col0 + lh;
    const float bv = bias[col];
    const int rbase = row0 + hsel * 8;

    if (full) {
        // branch-free: one base pointer, 8 constant-offset stores
        float* zp = z + (size_t)rbase * OUT_DIM + col;
        #pragma unroll
        for (int i = 0; i < 8; ++i) {
            float v = acc[i] + bv;
            if (isinf(v)) v = BIG;
            zp[(size_t)i * OUT_DIM] = v;
        }
    } else {
        #pragma unroll
        for (int i = 0; i < 8; ++i) {
            int r = rbase + i;
            if (r < n) {
                float v = acc[i] + bv;
                if (isinf(v)) v = BIG;
                z[(size_t)r * OUT_DIM + col] = v;
            }
        }
    }
}

// ---------------------------------------------------------------------------
// Kernel 2: per-node attention scores  s_src[n] = z[n].w_src, s_dst[n] = z[n].w_dst
// One wave32 per node; each lane handles 4 features (float4), shfl reduction.
// ---------------------------------------------------------------------------
__global__ __launch_bounds__(256) void scores_kernel(
    const float* __restrict__ z, const float* __restrict__ attn_w,
    float* __restrict__ s_src, float* __restrict__ s_dst, int n)
{
    const int lane = threadIdx.x & 31;
    const int wave = threadIdx.x >> 5;
    const int node = blockIdx.x * 8 + wave;
    if (node >= n) return;

    const float4 zv = ((const float4*)z)[(size_t)node * 32 + lane];
    const float4 ws = ((const float4*)attn_w)[lane];          // w_src
    const float4 wd = ((const float4*)attn_w)[32 + lane];     // w_dst

    float ss = zv.x * ws.x + zv.y * ws.y + zv.z * ws.z + zv.w * ws.w;
    float sd = zv.x * wd.x + zv.y * wd.y + zv.z * wd.z + zv.w * wd.w;

    #pragma unroll
    for (int off = 16; off > 0; off >>= 1) {
        ss += __shfl_down(ss, off, 32);
        sd += __shfl_down(sd, off, 32);
    }
    if (lane == 0) { s_src[node] = ss; s_dst[node] = sd; }
}

// ---------------------------------------------------------------------------
// atomic float max via int-bitpattern trick (init target with -inf)
// ---------------------------------------------------------------------------
__device__ __forceinline__ void atomicMaxF32(float* addr, float val) {
    if (val >= 0.0f) atomicMax((int*)addr, __float_as_int(val));
    else             atomicMin((unsigned int*)addr, __float_as_uint(val));
}

// ---------------------------------------------------------------------------
// Kernel 3: per-edge raw score e = leaky_relu(s_src[src]+s_dst[dst]+b), seg-max
// ---------------------------------------------------------------------------
__global__ __launch_bounds__(256) void edge_score_kernel(
    const int* __restrict__ src, const int* __restrict__ dst,
    const float* __restrict__ s_src, const float* __restrict__ s_dst,
    const float* __restrict__ attn_b, float* __restrict__ e_buf,
    float* __restrict__ m, int nE)
{
    int t = blockIdx.x * blockDim.x + threadIdx.x;
    if (t >= nE) return;
    float a = s_src[src[t]] + s_dst[dst[t]] + attn_b[0];
    if (isinf(a)) a = BIG;
    float e = (a > 0.0f) ? a : 0.01f * a;   // leaky_relu slope 0.01
    e_buf[t] = e;
    atomicMaxF32(&m[dst[t]], e);
}

// ---------------------------------------------------------------------------
// Kernel 4: ex = exp(e - m[dst]); denom[dst] += ex; bsum[dst] += sigma
// ---------------------------------------------------------------------------
__global__ __launch_bounds__(256) void edge_exp_kernel(
    const int* __restrict__ dst, const float* __restrict__ e_buf,
    const float* __restrict__ sigma, const float* __restrict__ m,
    float* __restrict__ ex_buf, float* __restrict__ denom,
    float* __restrict__ bsum, int nE)
{
    int t = blockIdx.x * blockDim.x + threadIdx.x;
    if (t >= nE) return;
    int d = dst[t];
    float ex = expf(e_buf[t] - m[d]);
    ex_buf[t] = ex;
    atomicAdd(&denom[d], ex);
    atomicAdd(&bsum[d], sigma[t]);
}

// ---------------------------------------------------------------------------
// Kernel 5: scatter  h_new[dst] += (alpha*beta) * z[src]
// One wave32 per edge, each lane does 4 features via float4 + 4 atomicAdds.
// ---------------------------------------------------------------------------
__global__ __launch_bounds__(256) void edge_scatter_kernel(
    const int* __restrict__ src, const int* __restrict__ dst,
    const float* __restrict__ sigma, const float* __restrict__ ex_buf,
    const float* __restrict__ denom, const float* __restrict__ bsum,
    const float* __restrict__ z, float* __restrict__ h_new, int nE)
{
    int gid  = blockIdx.x * blockDim.x + threadIdx.x;
    int edge = gid >> 5;
    int lane = gid & 31;
    if (edge >= nE) return;

    int d = dst[edge];
    int s = src[edge];
    float alpha = ex_buf[edge] / denom[d];
    float beta  = sigma[edge] / (bsum[d] + EPS_BETA);
    float coef  = alpha * beta;

    float4 zv = ((const float4*)z)[(size_t)s * 32 + lane];
    float* out = h_new + (size_t)d * OUT_DIM + lane * 4;
    atomicAdd(out + 0, coef * zv.x);
    atomicAdd(out + 1, coef * zv.y);
    atomicAdd(out + 2, coef * zv.z);
    atomicAdd(out + 3, coef * zv.w);
}

// ---------------------------------------------------------------------------
// Kernel 6: BN partial stats (sum, sumsq per feature), coalesced row sweeps
// ---------------------------------------------------------------------------
__global__ __launch_bounds__(128) void bn_stats_kernel(
    const float* __restrict__ h_new, float* __restrict__ fsum,
    float* __restrict__ fsumsq, int n)
{
    const int f = threadIdx.x;      // feature 0..127
    float acc = 0.0f, accsq = 0.0f;
    for (int row = blockIdx.x; row < n; row += gridDim.x) {
        float x = h_new[(size_t)row * OUT_DIM + f];
        if (isinf(x)) x = BIG;
        acc += x;
        accsq += x * x;
    }
    atomicAdd(&fsum[f], acc);
    atomicAdd(&fsumsq[f], accsq);
}

__global__ __launch_bounds__(128) void bn_finalize_kernel(
    const float* __restrict__ fsum, const float* __restrict__ fsumsq,
    float* __restrict__ mu, float* __restrict__ rstd, int n)
{
    int f = threadIdx.x;
    float m  = fsum[f] / (float)n;
    float v  = fsumsq[f] / (float)n - m * m;
    mu[f]   = m;
    rstd[f] = rsqrtf(v + EPS_BN);
}

// ---------------------------------------------------------------------------
// Kernel 7: out = elu((h_new - mu) * rstd * gamma + beta)
// ---------------------------------------------------------------------------
__global__ __launch_bounds__(256) void bn_elu_kernel(
    const float* __restrict__ h_new, const float* __restrict__ mu,
    const float* __restrict__ rstd, const float* __restrict__ gamma,
    const float* __restrict__ beta, float* __restrict__ out, int total)
{
    int i = blockIdx.x * blockDim.x + threadIdx.x;
    if (i >= total) return;
    int f = i & (OUT_DIM - 1);
    float x = h_new[i];
    if (isinf(x)) x = BIG;
    float y = (x - mu[f]) * rstd[f] * gamma[f] + beta[f];
    out[i] = (y > 0.0f) ? y : (expf(y) - 1.0f);   // ELU, alpha=1
}

// ---------------------------------------------------------------------------
extern "C" void kernel_launch(void* const* d_in, const int* in_sizes, int n_in,
                              void* d_out, int out_size, void* d_ws, size_t ws_size,
                              hipStream_t stream)
{
    const float* h        = (const float*)d_in[0];
    const int*   src      = (const int*)  d_in[1];
    const int*   dst      = (const int*)  d_in[2];
    const float* sigma    = (const float*)d_in[3];
    const float* fc_w     = (const float*)d_in[4];
    const float* fc_b     = (const float*)d_in[5];
    const float* attn_w   = (const float*)d_in[6];
    const float* attn_b   = (const float*)d_in[7];
    const float* bn_gamma = (const float*)d_in[8];
    const float* bn_beta  = (const float*)d_in[9];
    float*       out      = (float*)d_out;

    const int n  = in_sizes[0] / IN_DIM;   // 50000 nodes
    const int nE = in_sizes[1];            // 800000 edges
    const int rn = (n + 3) & ~3;           // 16B-aligned node-array length

    // workspace layout (floats)
    float* ws = (float*)d_ws;
    size_t off = 0;
    float* z      = ws + off; off += (size_t)n * OUT_DIM;
    float* h_new  = ws + off; off += (size_t)n * OUT_DIM;
    float* e_buf  = ws + off; off += (size_t)((nE + 3) & ~3);
    float* ex_buf = ws + off; off += (size_t)((nE + 3) & ~3);
    float* s_src  = ws + off; off += rn;
    float* s_dst  = ws + off; off += rn;
    float* m      = ws + off; off += rn;
    float* denom  = ws + off; off += rn;   // zero block starts here
    float* bsum   = ws + off; off += rn;
    float* fsum   = ws + off; off += 128;
    float* fsumsq = ws + off; off += 128;
    float* mu     = ws + off; off += 128;
    float* rstd   = ws + off; off += 128;

    // --- init ---
    {
        int zc = 2 * rn + 256;             // denom, bsum, fsum, fsumsq (contiguous)
        fill_kernel<<<(zc + 255) / 256, 256, 0, stream>>>(denom, 0.0f, zc);
        fill_kernel<<<(rn + 255) / 256, 256, 0, stream>>>(m, -INFINITY, rn);
        int hc = n * OUT_DIM;
        fill_kernel<<<(hc + 255) / 256, 256, 0, stream>>>(h_new, 0.0f, hc);
    }

    // --- 1: GEMM (WMMA fp32) ---
    gemm_bias_kernel<<<(n + 15) / 16, 256, 0, stream>>>(h, fc_w, fc_b, z, n);

    // --- 2: per-node scores ---
    scores_kernel<<<(n + 7) / 8, 256, 0, stream>>>(z, attn_w, s_src, s_dst, n);

    // --- 3: edge scores + segment max ---
    edge_score_kernel<<<(nE + 255) / 256, 256, 0, stream>>>(
        src, dst, s_src, s_dst, attn_b, e_buf, m, nE);

    // --- 4: exp + denom/bsum segment sums ---
    edge_exp_kernel<<<(nE + 255) / 256, 256, 0, stream>>>(
        dst, e_buf, sigma, m, ex_buf, denom, bsum, nE);

    // --- 5: weighted scatter-add of z[src] ---
    {
        long long tot = (long long)nE * 32;
        edge_scatter_kernel<<<(int)((tot + 255) / 256), 256, 0, stream>>>(
            src, dst, sigma, ex_buf, denom, bsum, z, h_new, nE);
    }

    // --- 6: BN stats ---
    bn_stats_kernel<<<256, 128, 0, stream>>>(h_new, fsum, fsumsq, n);
    bn_finalize_kernel<<<1, 128, 0, stream>>>(fsum, fsumsq, mu, rstd, n);

    // --- 7: normalize + ELU ---
    {
        int total = n * OUT_DIM;
        bn_elu_kernel<<<(total + 255) / 256, 256, 0, stream>>>(
            h_new, mu, rstd, bn_gamma, bn_beta, out, total);
    }
}